// FuncTT_36301063586442
// MI455X (gfx1250) — compile-verified
//
#include <hip/hip_runtime.h>
#include <math.h>

// TT functional evaluation on gfx1250 using V_WMMA_F32_16X16X4_F32.
// B=262144 points, 8 cores, DIM=32 basis, RANK=16, D_OUT=8.
//
// Per stage (transposed GEMM): res^T[k',b] += core[i,j,k'] * (res[b,i]*phi[b,j])
//   A (16x4, M=k', K=(i,j)) : batch-invariant core fragments staged in LDS,
//                             swizzled so each lane reads one ds_load_b64
//                             (pairs merge into ds_load_2addr_b64).
//   B (4x16, K=(i,j), N=b)  : built in registers (1 v_pk_mul_f32 per chunk).
//   C/D (16x16)             : res tile; D layout feeds the next stage directly.
// Cores are double-buffered in LDS (2 x 32 KB): stage s computes from buf[s&1]
// while all threads stage core s+1 into buf[(s+1)&1]; one barrier per stage.

typedef __attribute__((ext_vector_type(2)))  float v2f;
typedef __attribute__((ext_vector_type(8)))  float v8f;
typedef __attribute__((ext_vector_type(16))) float v16f;

#define B_TOTAL 262144
#define WAVES_PER_BLOCK 8
#define THREADS (WAVES_PER_BLOCK * 32)
#define LDS_ELEMS 8192  // largest core: 16*32*16 floats = 32 KB per buffer

// Copy core [R][32][C] from global into LDS with swizzle:
//   dst[i*512 + (j>>2)*64 + kp*4 + (j&3)] = core[i][j][kp]
// A WMMA K-chunk c (i=c>>3, jgroup=c&7) for lane (kp,khalf) is then the
// contiguous 8-byte pair at element offset c*64 + kp*4 + 2*khalf.
// For C==8 (last core) the kp=8..15 slots are zero-padded.
// Fully unrolled (compile-time trip count) so global loads issue in a burst.
template <int R, int C>
__device__ __forceinline__ void load_core_swizzled(float* dst, const float* __restrict__ src,
                                                   int tid) {
  constexpr int total = R * 32 * C;
  constexpr int iters = (total + THREADS - 1) / THREADS;
#pragma unroll
  for (int k = 0; k < iters; ++k) {
    int g  = tid + k * THREADS;
    int kp = g % C;
    int j  = (g / C) & 31;
    int i  = g / (32 * C);
    dst[i * 512 + (j >> 2) * 64 + kp * 4 + (j & 3)] = src[g];
  }
  if (C == 8) {
#pragma unroll
    for (int k = 0; k < iters; ++k) {
      int g  = tid + k * THREADS;
      int kp = 8 + (g & 7);
      int j  = (g >> 3) & 31;
      int i  = g >> 8;
      dst[i * 512 + (j >> 2) * 64 + kp * 4 + (j & 3)] = 0.0f;
    }
  }
}

// Orthonormal cosine basis phi_j(x) = (j ? sqrt2 : 1) * cos(j*x) via the
// Chebyshev recurrence (exactly one v_cos per point per dim), rolled through
// two scalars (pure SSA registers -> no private arrays).
// Returns the 16 values this lane's K-half needs:
//   q[2g+s] = phi[4g + 2*khalf + s], g = 0..7, s = 0..1.
__device__ __forceinline__ v16f compute_q(float x, bool kh) {
  const float S = 1.41421356237309504880f;
  float a = 1.0f;                // c_{j-1}
  float b = __builtin_cosf(x);   // c_j
  const float t2 = 2.0f * b;     // 2*cos(x)
  v16f q;
  {
    float p0 = 1.0f;
    float p1 = S * b;
    float n = t2 * b - a; a = b; b = n;  // c2
    float p2 = S * b;
    n = t2 * b - a; a = b; b = n;        // c3
    float p3 = S * b;
    q[0] = kh ? p2 : p0;
    q[1] = kh ? p3 : p1;
  }
#pragma unroll
  for (int g = 1; g < 8; ++g) {
    float n = t2 * b - a; a = b; b = n; float p0 = S * b;  // c_{4g}
    n = t2 * b - a; a = b; b = n;       float p1 = S * b;  // c_{4g+1}
    n = t2 * b - a; a = b; b = n;       float p2 = S * b;  // c_{4g+2}
    n = t2 * b - a; a = b; b = n;       float p3 = S * b;  // c_{4g+3}
    q[2 * g]     = kh ? p2 : p0;
    q[2 * g + 1] = kh ? p3 : p1;
  }
  return q;
}

// Expand the D-fragment res (res[v] = res[b][v + 8*khalf]) into all 16
// res[b][i] values per lane: 8 shuffles + 16 selects, hoisted per stage.
__device__ __forceinline__ v16f expand_res(const v8f& res, bool kh) {
  v16f full;
#pragma unroll
  for (int v = 0; v < 8; ++v) {
    float own = res[v];
    float oth = __shfl_xor(own, 16, 32);
    full[v]     = kh ? oth : own;  // i = v   lives in khalf==0 lanes
    full[v + 8] = kh ? own : oth;  // i = v+8 lives in khalf==1 lanes
  }
  return full;
}

// One full K=512 contraction: 16 i-slices x 8 j-groups = 128 WMMAs.
__device__ __forceinline__ v8f stage_gemm(const float* Ab, const v16f& q, const v16f& full) {
  v8f acc = {};
#pragma unroll
  for (int i = 0; i < 16; ++i) {
    float r = full[i];
    v2f rr = {r, r};
#pragma unroll
    for (int jg = 0; jg < 8; ++jg) {
      v2f a = *(const v2f*)(Ab + (i * 8 + jg) * 64);
      v2f qp = {q[2 * jg], q[2 * jg + 1]};
      v2f bb = qp * rr;
      acc = __builtin_amdgcn_wmma_f32_16x16x4_f32(false, a, false, bb, (short)0, acc,
                                                  false, false);
    }
  }
  return acc;
}

__global__ __launch_bounds__(THREADS)
void tt_eval_kernel(const float* __restrict__ x,
                    const float* __restrict__ c0, const float* __restrict__ c1,
                    const float* __restrict__ c2, const float* __restrict__ c3,
                    const float* __restrict__ c4, const float* __restrict__ c5,
                    const float* __restrict__ c6, const float* __restrict__ c7,
                    float* __restrict__ out) {
  __shared__ __align__(16) float lds[2][LDS_ELEMS];
  const int  tid  = threadIdx.x;
  const int  lane = tid & 31;
  const int  wave = tid >> 5;
  const int  kp   = lane & 15;         // batch point within tile (N) / k' row (A)
  const bool kh   = (lane >> 4) != 0;  // which K-half of the 4-wide WMMA chunk
  const int  base = (blockIdx.x * WAVES_PER_BLOCK + wave) * 16;
  const int  bpt  = base + kp;         // batch point this lane serves

  // Per-lane A base inside each swizzled LDS buffer; chunk offsets are
  // compile-time constants -> DS immediate offsets.
  const int laneoff = kp * 4 + (kh ? 2 : 0);
  const float* A0 = &lds[0][laneoff];
  const float* A1 = &lds[1][laneoff];

  // Prime both buffers: core0 -> buf0, core1 -> buf1.
  load_core_swizzled<1, 16>(lds[0], c0, tid);
  load_core_swizzled<16, 16>(lds[1], c1, tid);
  __syncthreads();

  // ----- stage 0: core0 [1,32,16], res starts as all-ones (8 chunks) -----
  v16f q = compute_q(x[bpt * 8 + 0], kh);
  v8f acc = {};
#pragma unroll
  for (int jg = 0; jg < 8; ++jg) {
    v2f a = *(const v2f*)(A0 + jg * 64);
    v2f bb = {q[2 * jg], q[2 * jg + 1]};
    acc = __builtin_amdgcn_wmma_f32_16x16x4_f32(false, a, false, bb, (short)0, acc,
                                                false, false);
  }
  v8f res = acc;
  __syncthreads();  // buf0 reads retired -> buf0 free for core2

  // ----- stages 1..6: cores [16,32,16], K = 512 -----
#pragma unroll 1
  for (int s = 1; s < 7; ++s) {
    // Stage core s+1 into the buffer freed at the previous barrier, while
    // computing stage s from the other buffer.
    float* nbuf = lds[(s + 1) & 1];
    if (s < 6) {
      const float* cn = (s == 1) ? c2 : (s == 2) ? c3 : (s == 3) ? c4
                       : (s == 4) ? c5 : c6;
      load_core_swizzled<16, 16>(nbuf, cn, tid);
    } else {
      load_core_swizzled<16, 8>(nbuf, c7, tid);
    }

    q = compute_q(x[bpt * 8 + s], kh);
    v16f full = expand_res(res, kh);
    res = stage_gemm((s & 1) ? A1 : A0, q, full);
    __syncthreads();  // retire reads of buf[s&1] and writes of buf[(s+1)&1]
  }

  // ----- stage 7: core7 [16,32,8], zero-padded to 16 output rows -----
  q = compute_q(x[bpt * 8 + 7], kh);
  v16f full = expand_res(res, kh);
  acc = stage_gemm(A1, q, full);

  // Output: lanes 0..15 hold out[b = base+kp][k' = v], v = 0..7.
  if (!kh) {
    float* o = out + (size_t)bpt * 8;
#pragma unroll
    for (int v = 0; v < 8; ++v) o[v] = acc[v];
  }
}

extern "C" void kernel_launch(void* const* d_in, const int* in_sizes, int n_in,
                              void* d_out, int out_size, void* d_ws, size_t ws_size,
                              hipStream_t stream) {
  // d_in[0] = t (unused scalar), d_in[1] = x [B,8] f32, d_in[2..9] = cores 0..7
  const float* x  = (const float*)d_in[1];
  const float* c0 = (const float*)d_in[2];
  const float* c1 = (const float*)d_in[3];
  const float* c2 = (const float*)d_in[4];
  const float* c3 = (const float*)d_in[5];
  const float* c4 = (const float*)d_in[6];
  const float* c5 = (const float*)d_in[7];
  const float* c6 = (const float*)d_in[8];
  const float* c7 = (const float*)d_in[9];
  float* out = (float*)d_out;

  int tiles  = B_TOTAL / 16;             // 16384 wave-tiles
  int blocks = tiles / WAVES_PER_BLOCK;  // 2048 blocks of 8 waves
  tt_eval_kernel<<<blocks, THREADS, 0, stream>>>(x, c0, c1, c2, c3, c4, c5, c6, c7, out);
}